// Resample_77970836291694
// MI455X (gfx1250) — compile-verified
//
#include <hip/hip_runtime.h>
#include <stdint.h>

#ifndef __has_builtin
#define __has_builtin(x) 0
#endif

typedef float v4f __attribute__((ext_vector_type(4)));
typedef int   v4i __attribute__((ext_vector_type(4)));

#define AS1 __attribute__((address_space(1)))
#define AS3 __attribute__((address_space(3)))

#if __has_builtin(__builtin_amdgcn_global_load_async_to_lds_b128)
#define HAVE_ASYNC_LDS 1
#else
#define HAVE_ASYNC_LDS 0
#endif

__device__ __forceinline__ void upd(float v, unsigned i, float& bv, unsigned& bi) {
    if (v > bv) { bv = v; bi = i; }   // strict > keeps earliest index in ascending scan
}

__device__ __forceinline__ void upd4(v4f v, unsigned i0, float& bv, unsigned& bi) {
    upd(v.x, i0 + 0u, bv, bi);
    upd(v.y, i0 + 1u, bv, bi);
    upd(v.z, i0 + 2u, bv, bi);
    upd(v.w, i0 + 3u, bv, bi);
}

#if HAVE_ASYNC_LDS
__device__ __forceinline__ void async_copy_b128(const float* gsrc, v4f* ldst) {
    // One instruction per wave: 32 lanes x 16B = 512B global -> LDS, tracked on ASYNCcnt.
    __builtin_amdgcn_global_load_async_to_lds_b128((AS1 v4i*)gsrc, (AS3 v4i*)ldst,
                                                   /*offset=*/0, /*cpol=*/0);
}

template <int N>
__device__ __forceinline__ void wait_asynccnt() {
#if __has_builtin(__builtin_amdgcn_s_wait_asynccnt)
    __builtin_amdgcn_s_wait_asynccnt(N);
#else
    asm volatile("s_wait_asynccnt %0" :: "n"(N));
#endif
    asm volatile("" ::: "memory");   // keep LDS reads below the wait
}
#endif

__global__ __launch_bounds__(256, 1) void Resample_77970836291694_kernel(
    const float* __restrict__ target, float* __restrict__ out)
{
    constexpr int H = 256, W = 256, HW = H * W;     // 65536 elems per (b,c)
    constexpr int OH = 64, OW = 64, OHW = OH * OW;  // 4096 out elems per (b,c)
    constexpr int RH = H / OH, RW = W / OW;         // 4, 4

    const int bc  = blockIdx.x;
    const int tid = threadIdx.x;
    const float* __restrict__ base = target + (size_t)bc * HW;

    float    bv = -__builtin_inff();
    unsigned bi = 0u;

#if HAVE_ASYNC_LDS
    // ---- async global->LDS streaming pipeline (per-wave strips, 4-deep) ----
    const int wave = tid >> 5;
    const int lane = tid & 31;
    __shared__ v4f abuf[8][4][32];                       // 16 KB staging
    const float*   wbase = base + wave * 8192;           // 8192 floats per wave
    const unsigned ibase = (unsigned)(wave * 8192 + lane * 4);

    // prime: 3 stages in flight
    async_copy_b128(wbase + 0 * 128 + lane * 4, &abuf[wave][0][lane]);
    async_copy_b128(wbase + 1 * 128 + lane * 4, &abuf[wave][1][lane]);
    async_copy_b128(wbase + 2 * 128 + lane * 4, &abuf[wave][2][lane]);

    for (int s = 0; s < 61; ++s) {
        async_copy_b128(wbase + (s + 3) * 128 + lane * 4, &abuf[wave][(s + 3) & 3][lane]);
        wait_asynccnt<3>();                              // stage s retired (in-order)
        v4f v = abuf[wave][s & 3][lane];
        upd4(v, ibase + (unsigned)s * 128u, bv, bi);
        asm volatile("" ::: "memory");                   // ds_load stays before next issue
    }
    wait_asynccnt<2>();
    upd4(abuf[wave][61 & 3][lane], ibase + 61u * 128u, bv, bi);
    wait_asynccnt<1>();
    upd4(abuf[wave][62 & 3][lane], ibase + 62u * 128u, bv, bi);
    wait_asynccnt<0>();
    upd4(abuf[wave][63 & 3][lane], ibase + 63u * 128u, bv, bi);
#else
    // ---- fallback: direct non-temporal b128 streaming loads ----
    const v4f* __restrict__ vbase = (const v4f*)base;
    #pragma unroll 4
    for (int s = 0; s < 64; ++s) {
        v4f v = __builtin_nontemporal_load(vbase + s * 256 + tid);
        upd4(v, (unsigned)((s * 256 + tid) * 4), bv, bi);
    }
#endif

    // ---- wave32 butterfly reduction (value, lowest-index tie-break) ----
    #pragma unroll
    for (int off = 16; off > 0; off >>= 1) {
        float    ov = __shfl_xor(bv, off, 32);
        unsigned oi = __shfl_xor(bi, off, 32);
        if (ov > bv || (ov == bv && oi < bi)) { bv = ov; bi = oi; }
    }

    // ---- cross-wave reduction via LDS ----
    __shared__ float    s_v[8];
    __shared__ unsigned s_i[8];
    if ((tid & 31) == 0) { s_v[tid >> 5] = bv; s_i[tid >> 5] = bi; }
    __syncthreads();

    float    fv = s_v[0];
    unsigned fi = s_i[0];
    #pragma unroll
    for (int w = 1; w < 8; ++w) {
        float    v = s_v[w];
        unsigned i = s_i[w];
        if (v > fv || (v == fv && i < fi)) { fv = v; fi = i; }
    }

    const unsigned row = (fi / (unsigned)W) / (unsigned)RH;
    const unsigned col = (fi % (unsigned)W) / (unsigned)RW;
    const unsigned p   = row * (unsigned)OW + col;

    // ---- one-shot tile write: each element written exactly once, no fence needed ----
    float* __restrict__ otile = out + (size_t)bc * OHW;
    #pragma unroll
    for (int j = 0; j < 4; ++j) {
        const unsigned e = (unsigned)tid * 16u + (unsigned)j * 4u;
        v4f z;
        z.x = (e + 0u == p) ? 1.0f : 0.0f;
        z.y = (e + 1u == p) ? 1.0f : 0.0f;
        z.z = (e + 2u == p) ? 1.0f : 0.0f;
        z.w = (e + 3u == p) ? 1.0f : 0.0f;
        __builtin_nontemporal_store(z, (v4f*)(otile + e));
    }
}

extern "C" void kernel_launch(void* const* d_in, const int* in_sizes, int n_in,
                              void* d_out, int out_size, void* d_ws, size_t ws_size,
                              hipStream_t stream) {
    (void)n_in; (void)out_size; (void)d_ws; (void)ws_size;
    const float* target = (const float*)d_in[0];
    float*       out    = (float*)d_out;
    // B*C heatmaps of 256x256; one workgroup (8 wave32s) per heatmap.
    const int nbc = in_sizes[0] / (256 * 256);   // 2048 for the reference shapes
    if (nbc <= 0) return;
    Resample_77970836291694_kernel<<<nbc, 256, 0, stream>>>(target, out);
}